// GTMaskedQueryAndGroup_55997783605459
// MI455X (gfx1250) — compile-verified
//
#include <hip/hip_runtime.h>
#include <hip/hip_bf16.h>

typedef __attribute__((ext_vector_type(2))) float v2f;
typedef __attribute__((ext_vector_type(8))) float v8f;

// Problem constants (match setup_inputs)
constexpr int BATCH   = 2;
constexpr int NQ      = 4608;
constexpr int NS      = 8192;
constexpr int KSAMP   = 16;
constexpr int GROUPS  = 9;
constexpr int NPOINT  = NQ / GROUPS;      // 512
constexpr int CV      = 512;
constexpr int DV      = CV / 4;           // 128
constexpr int NLG     = 8;
constexpr int NLCH    = DV / NLG;         // 16
constexpr float R2    = 0.1f * 0.1f;

// Output region sizes (float elements), in tuple return order
constexpr size_t N_LF  = (size_t)BATCH * GROUPS * (3 + DV) * NPOINT * KSAMP; // 19316736
constexpr size_t N_NLF = (size_t)BATCH * NLG * (3 + NLCH) * NPOINT * KSAMP;  // 2490368
constexpr size_t N_MSK = (size_t)BATCH * NQ * KSAMP;                         // 147456

// ---------------------------------------------------------------------------
// Kernel 1: ordered ball query, 16x16 distance tiles via V_WMMA_F32_16X16X4_F32
// One wave32 per tile of 16 queries; loops over support in tiles of 16.
// ---------------------------------------------------------------------------
__global__ __launch_bounds__(32) void ball_query_kernel(
    const float* __restrict__ qxyz, const float* __restrict__ sxyz,
    const unsigned char* __restrict__ qmask, const unsigned char* __restrict__ smask,
    int* __restrict__ idx_out, float* __restrict__ mask_out)
{
    const int tile  = blockIdx.x;                 // 0 .. BATCH*NQ/16-1
    const int b     = tile / (NQ / 16);
    const int qbase = (tile % (NQ / 16)) * 16;
    const int lane  = threadIdx.x;                // wave32
    const int r16   = lane & 15;
    const bool hi   = lane >= 16;

    // Each lane owns query row (lane&15): both half-waves duplicate the rows.
    const float* qp = qxyz + ((size_t)b * NQ + qbase + r16) * 3;
    const float qx = qp[0], qy = qp[1], qz = qp[2];
    const float qn = qx * qx + qy * qy + qz * qz;
    const int qval = qmask[(size_t)b * NQ + qbase + r16] ? 1 : 0;

    // A matrix (16x4 f32): lanes 0-15 -> {K0,K1}={x,y}; lanes 16-31 -> {K2,K3}={z,0}
    v2f A;
    A.x = hi ? qz : qx;
    A.y = hi ? 0.0f : qy;

    // Collection state: lanes 0..15 own query rows 0..15
    int mycnt    = (lane < 16) ? 0 : KSAMP;   // idle lanes report "done"
    int firstidx = 0;

    for (int sbase = 0; sbase < NS; sbase += 16) {
        // Each lane loads support col (lane&15)
        const int col   = sbase + r16;
        const float* sp = sxyz + ((size_t)b * NS + col) * 3;
        const float sx = sp[0], sy = sp[1], sz = sp[2];
        const float sn = sx * sx + sy * sy + sz * sz;
        const int sval = smask[(size_t)b * NS + col] ? 1 : 0;

        // B matrix (4x16 f32): lanes 0-15 rows K0,K1 = x,y; lanes 16-31 rows K2,K3 = z,0
        v2f Bm;
        Bm.x = hi ? sz : sx;
        Bm.y = hi ? 0.0f : sy;

        v8f acc = {};
#if __has_builtin(__builtin_amdgcn_wmma_f32_16x16x4_f32)
        acc = __builtin_amdgcn_wmma_f32_16x16x4_f32(
            /*neg_a=*/false, A, /*neg_b=*/false, Bm,
            /*c_mod=*/(short)0, acc, /*reuse_a=*/false, /*reuse_b=*/false);
#else
        // Fallback: compute the same C layout with shuffles (should not trigger)
        #pragma unroll
        for (int v = 0; v < 8; ++v) {
            const int r = v + (hi ? 8 : 0);
            const float ax = __shfl(qx, r, 32);
            const float ay = __shfl(qy, r, 32);
            const float az = __shfl(qz, r, 32);
            acc[v] = ax * sx + ay * sy + az * sz;
        }
#endif
        // C layout: VGPR v, lanes 0-15 -> (row v, col=lane); lanes 16-31 -> (row v+8, col=lane-16)
        unsigned int ball[8];
        #pragma unroll
        for (int v = 0; v < 8; ++v) {
            const int r = v + (hi ? 8 : 0);
            const float qn_r  = __shfl(qn, r, 32);
            const int   qv_r  = __shfl(qval, r, 32);
            const float d2    = qn_r + sn - 2.0f * acc[v];
            const bool valid  = (d2 < R2) && (sval != 0) && (qv_r != 0);
            ball[v] = (unsigned int)__ballot(valid);   // bits[15:0]=row v, bits[31:16]=row v+8
        }

        if (lane < 16) {
            // Row owned by this lane
            unsigned int bm = 0;
            #pragma unroll
            for (int v = 0; v < 8; ++v) bm = ((lane & 7) == v) ? ball[v] : bm;
            unsigned int m = (bm >> ((lane & 8) ? 16 : 0)) & 0xFFFFu;
            while (m && mycnt < KSAMP) {
                const int c = __builtin_ctz(m);
                m &= m - 1u;
                const int sidx = sbase + c;
                if (mycnt == 0) firstidx = sidx;
                idx_out[((size_t)b * NQ + qbase + lane) * KSAMP + mycnt] = sidx;
                ++mycnt;
            }
        }
        // Uniform early exit keeps EXEC full for the next WMMA
        if (__ballot(mycnt >= KSAMP) == 0xFFFFFFFFull) break;
    }

    if (lane < 16) {
        const size_t qrow = (size_t)b * NQ + qbase + lane;
        for (int s = mycnt; s < KSAMP; ++s) idx_out[qrow * KSAMP + s] = firstidx;
        for (int s = 0; s < KSAMP; ++s)
            mask_out[qrow * KSAMP + s] = (s < mycnt) ? 1.0f : 0.0f;
    }
}

// ---------------------------------------------------------------------------
// Kernel 2: gather local features (B,9,131,512,16) + qk_out (B,64,NQ)
// One 256-thread block per (b,q).
// ---------------------------------------------------------------------------
__global__ __launch_bounds__(256) void gather_local_kernel(
    const float* __restrict__ qxyz, const float* __restrict__ sxyz,
    const float* __restrict__ value, const float* __restrict__ qk,
    const int* __restrict__ idx, float* __restrict__ lf_out, float* __restrict__ qk_out)
{
    const int bq = blockIdx.x;          // b*NQ + q
    const int b  = bq / NQ;
    const int q  = bq % NQ;
    const int g  = q / NPOINT;
    const int p  = q % NPOINT;
    const int tid = threadIdx.x;

    __shared__ int   sidx[KSAMP];
    __shared__ float sq[3];
    if (tid < KSAMP) sidx[tid] = idx[(size_t)bq * KSAMP + tid];
    if (tid < 3)     sq[tid]   = qxyz[(size_t)bq * 3 + tid];
    __syncthreads();

    // qk_out: 64 channels gathered at the first sample index
    const int i0 = sidx[0];
    if (tid < 64)
        qk_out[((size_t)b * 64 + tid) * NQ + q] = qk[((size_t)b * 64 + tid) * NS + i0];

    // local features: c in [0,131): c<3 -> (sxyz - qxyz), else localvalue channel c-3
    const int k  = tid & 15;
    const int si = sidx[k];
    for (int c = tid >> 4; c < 3 + DV; c += 16) {
        float val;
        if (c < 3) val = sxyz[((size_t)b * NS + si) * 3 + c] - sq[c];
        else       val = value[((size_t)b * CV + DV + (c - 3)) * NS + si];
        lf_out[((((size_t)b * GROUPS + g) * (3 + DV) + c) * NPOINT + p) * KSAMP + k] = val;
    }
}

// ---------------------------------------------------------------------------
// Kernel 3: top-16 of attention_centrality per (b, group); values staged in LDS
// ---------------------------------------------------------------------------
__global__ __launch_bounds__(256) void topk_kernel(
    const float* __restrict__ ac, int* __restrict__ idx_ac)
{
    const int bg  = blockIdx.x;         // 0 .. BATCH*NLG-1
    const int tid = threadIdx.x;
    __shared__ float vals[NS];          // 32 KB
    __shared__ float rv[256];
    __shared__ int   ri[256];

    const float* a = ac + (size_t)bg * NS;
    for (int i = tid; i < NS; i += 256) vals[i] = a[i];
    __syncthreads();

    for (int t = 0; t < KSAMP; ++t) {
        float bv = -INFINITY;
        int   bi = NS;
        for (int i = tid; i < NS; i += 256) {
            const float v = vals[i];
            if (v > bv || (v == bv && i < bi)) { bv = v; bi = i; }
        }
        rv[tid] = bv; ri[tid] = bi;
        __syncthreads();
        for (int s = 128; s > 0; s >>= 1) {
            if (tid < s) {
                if (rv[tid + s] > rv[tid] ||
                    (rv[tid + s] == rv[tid] && ri[tid + s] < ri[tid])) {
                    rv[tid] = rv[tid + s];
                    ri[tid] = ri[tid + s];
                }
            }
            __syncthreads();
        }
        if (tid == 0) {
            idx_ac[bg * KSAMP + t] = ri[0];
            vals[ri[0]] = -INFINITY;    // exclude from later rounds
        }
        __syncthreads();
    }
}

// ---------------------------------------------------------------------------
// Kernel 4: nonlocal features (B,8,19,512,16). Gathered tiles are p-invariant:
// stage once in LDS, then stream writes. Grid: (bj, p-chunk of 16).
// ---------------------------------------------------------------------------
__global__ __launch_bounds__(256) void gather_nonlocal_kernel(
    const float* __restrict__ qxyz, const float* __restrict__ sxyz,
    const float* __restrict__ value, const int* __restrict__ idx_ac,
    float* __restrict__ nlf_out)
{
    constexpr int PCHUNK = 16;
    const int bj = blockIdx.x / (NPOINT / PCHUNK);          // b*NLG + j
    const int pc = blockIdx.x % (NPOINT / PCHUNK);
    const int b  = bj / NLG;
    const int j  = bj % NLG;
    const int tid = threadIdx.x;

    __shared__ int   sidx[KSAMP];
    __shared__ float gx[3][KSAMP];
    __shared__ float gv[NLCH][KSAMP];
    if (tid < KSAMP) sidx[tid] = idx_ac[bj * KSAMP + tid];
    __syncthreads();
    if (tid < 3 * KSAMP) {
        const int c = tid >> 4, k = tid & 15;
        gx[c][k] = sxyz[((size_t)b * NS + sidx[k]) * 3 + c];
    }
    {   // 16 channels x 16 samples == 256 threads
        const int ch = tid >> 4, k = tid & 15;
        gv[ch][k] = value[((size_t)b * CV + j * NLCH + ch) * NS + sidx[k]];
    }
    __syncthreads();

    const int k    = tid & 15;
    const int psub = tid >> 4;                  // 0..15
    const int p    = pc * PCHUNK + psub;
    #pragma unroll
    for (int c = 0; c < 3 + NLCH; ++c) {
        float val;
        if (c < 3) val = gx[c][k] - qxyz[((size_t)b * NQ + p) * 3 + c];
        else       val = gv[c - 3][k];
        nlf_out[(((size_t)bj * (3 + NLCH) + c) * NPOINT + p) * KSAMP + k] = val;
    }
}

// ---------------------------------------------------------------------------
extern "C" void kernel_launch(void* const* d_in, const int* in_sizes, int n_in,
                              void* d_out, int out_size, void* d_ws, size_t ws_size,
                              hipStream_t stream)
{
    (void)in_sizes; (void)n_in; (void)out_size; (void)ws_size;

    const float*         qxyz  = (const float*)d_in[0];
    const float*         sxyz  = (const float*)d_in[1];
    const unsigned char* qmask = (const unsigned char*)d_in[2];
    const unsigned char* smask = (const unsigned char*)d_in[3];
    const float*         qk    = (const float*)d_in[4];
    const float*         value = (const float*)d_in[5];
    const float*         ac    = (const float*)d_in[6];

    float* out  = (float*)d_out;
    float* lf   = out;                    // (B,9,131,512,16)
    float* nlf  = lf + N_LF;              // (B,8,19,512,16)
    float* mk   = nlf + N_NLF;            // (B,NQ,16) as 0/1 floats
    float* qko  = mk + N_MSK;             // (B,64,NQ)

    int* idx    = (int*)d_ws;                          // (B,NQ,16)
    int* idx_ac = idx + (size_t)BATCH * NQ * KSAMP;    // (B,8,16)

    ball_query_kernel<<<BATCH * (NQ / 16), 32, 0, stream>>>(
        qxyz, sxyz, qmask, smask, idx, mk);
    topk_kernel<<<BATCH * NLG, 256, 0, stream>>>(ac, idx_ac);
    gather_local_kernel<<<BATCH * NQ, 256, 0, stream>>>(
        qxyz, sxyz, value, qk, idx, lf, qko);
    gather_nonlocal_kernel<<<BATCH * NLG * (NPOINT / 16), 256, 0, stream>>>(
        qxyz, sxyz, value, idx_ac, nlf);
}